// SimpleRNN_62173946577512
// MI455X (gfx1250) — compile-verified
//
#include <hip/hip_runtime.h>
#include <math.h>

// Problem constants (from reference): B=1024, T=500, F=64, H=4
#define B_ 1024
#define T_ 500
#define F_ 64
#define H_ 4

typedef __attribute__((ext_vector_type(2))) float v2f;
typedef __attribute__((ext_vector_type(8))) float v8f;

#if __has_builtin(__builtin_amdgcn_tanhf)
__device__ __forceinline__ float fast_tanh(float x) { return __builtin_amdgcn_tanhf(x); }
#else
__device__ __forceinline__ float fast_tanh(float x) { return tanhf(x); }
#endif

// ---------------------------------------------------------------------------
// Kernel 1: xp[t][b][h] = sum_f x[b][t][f] * Wx[f][h] + bias[h]
// GEMM [B*T x 64] x [64 x 4] via V_WMMA_F32_16X16X4_F32, K chained 16x.
// One 16-row M-tile per wave iteration; grid-stride over 32000 tiles.
// xp layout: [T][B][H] (8 MB -> resident in 192MB L2 for kernel 2).
// ---------------------------------------------------------------------------
__global__ void __launch_bounds__(256) rnn_proj_wmma(
    const float* __restrict__ x, const float* __restrict__ Wx,
    const float* __restrict__ bias, float* __restrict__ xp)
{
    const int lane  = threadIdx.x & 31;
    const int n     = lane & 15;       // N column of D / column of B
    const int hi    = lane >> 4;       // half-wave select
    const int khalf = hi ? 2 : 0;      // A: lanes 16-31 hold K=2,3

    // Preload B fragments: Wx is [64][4]; pad columns 4..15 with zero.
    // B 4x16 layout: VGPR0 = rows K=0 (lanes 0-15) / K=2 (lanes 16-31),
    //                VGPR1 = rows K=1 / K=3.
    v2f bfrag[16];
#pragma unroll
    for (int kk = 0; kk < 16; ++kk) {
        const int k0 = 4 * kk + khalf;
        float b0 = 0.0f, b1 = 0.0f;
        if (n < H_) {
            b0 = Wx[k0 * H_ + n];
            b1 = Wx[(k0 + 1) * H_ + n];
        }
        v2f t; t.x = b0; t.y = b1;
        bfrag[kk] = t;
    }
    const float biasn = (n < H_) ? bias[n] : 0.0f;

    const int nwaves = (gridDim.x * blockDim.x) >> 5;
    const int wave   = (blockIdx.x * blockDim.x + threadIdx.x) >> 5;
    const int ntiles = (B_ * T_) / 16;  // 32000, exact

    for (int tile = wave; tile < ntiles; tile += nwaves) {
        const int mbase = tile * 16;
        // A 16x4 layout: lane holds row M = mbase + (lane&15), K = khalf..khalf+1
        const float* xrow = x + (size_t)(mbase + n) * F_ + khalf;

        v8f c = {0.f, 0.f, 0.f, 0.f, 0.f, 0.f, 0.f, 0.f};
#pragma unroll
        for (int kk = 0; kk < 16; ++kk) {
            v2f a = *(const v2f*)(xrow + 4 * kk);  // 8B-aligned
            c = __builtin_amdgcn_wmma_f32_16x16x4_f32(
                    false, a, false, bfrag[kk], (short)0, c, false, false);
        }

        // D layout: VGPR v -> (M = mbase+v, N = lane) lanes 0-15,
        //                     (M = mbase+8+v, N = lane-16) lanes 16-31.
        if (n < H_) {
#pragma unroll
            for (int v = 0; v < 8; ++v) {
                const int m  = mbase + v + (hi ? 8 : 0); // global row = b*T + t
                const int bb = m / T_;
                const int tt = m - bb * T_;
                xp[(size_t)tt * (B_ * H_) + bb * H_ + n] = c[v] + biasn;
            }
        }
    }
}

// ---------------------------------------------------------------------------
// Kernel 2: sequential scan  h = tanh(xp_t + h @ Wh),  then sigmoid head.
// One lane per batch element (1024 lanes = 32 waves); h0..h3 in VGPRs.
// xp reads hit L2 (just written by kernel 1). Hardware v_tanh_f32 when
// available keeps the 500-step dependency chain short.
// ---------------------------------------------------------------------------
__global__ void __launch_bounds__(256) rnn_scan(
    const float* __restrict__ xp, const float* __restrict__ Wh,
    const float* __restrict__ Wd, const float* __restrict__ bd,
    float* __restrict__ out)
{
    const int b = blockIdx.x * blockDim.x + threadIdx.x;
    if (b >= B_) return;

    float w[16];
#pragma unroll
    for (int i = 0; i < 16; ++i) w[i] = Wh[i];  // uniform -> scalar loads

    float h0 = 0.f, h1 = 0.f, h2 = 0.f, h3 = 0.f;
    const float4* p = (const float4*)xp;  // [T][B] of float4 (H=4)

#pragma unroll 2
    for (int t = 0; t < T_; ++t) {
        const float4 pv = p[(size_t)t * B_ + b];  // load independent of h
        const float z0 = pv.x + h0 * w[0] + h1 * w[4] + h2 * w[8]  + h3 * w[12];
        const float z1 = pv.y + h0 * w[1] + h1 * w[5] + h2 * w[9]  + h3 * w[13];
        const float z2 = pv.z + h0 * w[2] + h1 * w[6] + h2 * w[10] + h3 * w[14];
        const float z3 = pv.w + h0 * w[3] + h1 * w[7] + h2 * w[11] + h3 * w[15];
        h0 = fast_tanh(z0);
        h1 = fast_tanh(z1);
        h2 = fast_tanh(z2);
        h3 = fast_tanh(z3);
    }

    const float z = h0 * Wd[0] + h1 * Wd[1] + h2 * Wd[2] + h3 * Wd[3] + bd[0];
    out[b] = 1.0f / (1.0f + __expf(-z));
}

// ---------------------------------------------------------------------------
extern "C" void kernel_launch(void* const* d_in, const int* in_sizes, int n_in,
                              void* d_out, int out_size, void* d_ws, size_t ws_size,
                              hipStream_t stream) {
    const float* x   = (const float*)d_in[0];  // [B,T,F]
    const float* Wx  = (const float*)d_in[1];  // [F,H]
    const float* Wh  = (const float*)d_in[2];  // [H,H]
    const float* bia = (const float*)d_in[3];  // [H]
    const float* Wd  = (const float*)d_in[4];  // [H,1]
    const float* bd  = (const float*)d_in[5];  // [1]

    float* xp = (float*)d_ws;  // needs T*B*H*4 = 8,192,000 bytes of scratch

    // Kernel 1: 512 blocks x 8 waves = 4096 waves; ~7.8 tiles each.
    rnn_proj_wmma<<<512, 256, 0, stream>>>(x, Wx, bia, xp);

    // Kernel 2: 1024 threads = 32 waves, one lane per batch element.
    rnn_scan<<<(B_ + 255) / 256, 256, 0, stream>>>(xp, Wh, Wd, bd, (float*)d_out);
}